// myRNN_7524782702668
// MI455X (gfx1250) — compile-verified
//
#include <hip/hip_runtime.h>
#include <cmath>

typedef __attribute__((ext_vector_type(2))) float v2f;
typedef __attribute__((ext_vector_type(8))) float v8f;

__device__ __forceinline__ float fast_tanh(float v) {
#if __has_builtin(__builtin_amdgcn_tanhf)
    return __builtin_amdgcn_tanhf(v);
#else
    return tanhf(v);
#endif
}

// Involution on the hidden-index storage order: swap bits 1 and 3.
// Chosen so that the D-layout half (bit3) of slot mu(q) equals the
// B-layout half (bit1) of slot q -> the timestep handoff is lane-local.
__device__ __forceinline__ int swap13(int v) {
    const int b1 = (v >> 1) & 1, b3 = (v >> 3) & 1;
    return (v & ~0xA) | (b1 << 3) | (b3 << 1);
}

// One wave32 per 16-batch-row tile. Computes the transposed recurrence
//   H^T_new = tanh(P W_hh * H^T + P XP^T)   (P = mu row-permutation)
// with V_WMMA_F32_16X16X4_F32. W_hh lives in loop-invariant A fragments;
// thanks to the mu-relabeling the D->B handoff between timesteps is a pure
// register rename: no LDS, no shuffles on the serial chain.
__global__ __launch_bounds__(32)
void rnn_scan_wmma_f32(const float* __restrict__ x,       // [B, T]
                       const float* __restrict__ hidden,  // [B, H]
                       const float* __restrict__ W_ih,    // [H]
                       const float* __restrict__ W_hh,    // [H, H]
                       const float* __restrict__ b_ih,    // [H]
                       const float* __restrict__ b_hh,    // [H]
                       const float* __restrict__ W_fc,    // [H]
                       const float* __restrict__ b_fc,    // [1]
                       float* __restrict__ out,           // [B*T]
                       float* __restrict__ h_out,         // [B, H]
                       int B, int T)
{
    constexpr int H = 32;

    const int lane = threadIdx.x;          // 0..31
    const int half = lane >> 4;            // 0 | 1
    const int l15  = lane & 15;
    const int b0   = blockIdx.x * 16;      // batch tile base

    // ---- loop-invariant A fragments: A[p][q] = W_hh[mu(p)][q] ----
    // A 16x4 f32 layout: lanes0-15 hold {K=k0,k0+1}, lanes16-31 {K=k0+2,k0+3}, M = lane&15.
    v2f a_frag[2][8];
#pragma unroll
    for (int mt = 0; mt < 2; ++mt) {
        const int mrow = 16 * mt + swap13(l15);     // logical W_hh row for this phys slot
#pragma unroll
        for (int kk = 0; kk < 8; ++kk) {
            const float* p = W_hh + mrow * H + 4 * kk + 2 * half;
            v2f a; a.x = p[0]; a.y = p[1];
            a_frag[mt][kk] = a;
        }
    }

    // ---- per-lane scalar constants (mu-relabeled) for C build / output / h_out ----
    // C/D layout: VGPR r, lanes0-15 -> p = 16*mt + r ; lanes16-31 -> p = 16*mt + r + 8.
    float wih[2][8], bia[2][8], wfc[2][8];
#pragma unroll
    for (int mt = 0; mt < 2; ++mt)
#pragma unroll
        for (int r = 0; r < 8; ++r) {
            const int hh = swap13(16 * mt + r + 8 * half);   // logical h-index of this slot
            wih[mt][r] = W_ih[hh];
            bia[mt][r] = b_ih[hh] + b_hh[hh];
            wfc[mt][r] = W_fc[hh];
        }
    const float bfc = b_fc[0];

    // ---- initial B fragments straight from global hidden (identity storage) ----
    // B 4x16 f32 layout: lanes0-15 {k0,k0+1}, lanes16-31 {k0+2,k0+3}, N = lane&15.
    v2f b_frag[8];
#pragma unroll
    for (int kk = 0; kk < 8; ++kk) {
        const float* p = hidden + (b0 + l15) * H + 4 * kk + 2 * half;
        v2f b; b.x = p[0]; b.y = p[1];
        b_frag[kk] = b;
    }

    const float* xrow = x   + (size_t)(b0 + l15) * T;  // lane's batch row of x
    float*       orow = out + (size_t)(b0 + l15) * T;  // lane's batch row of out

    const v8f zeroC = {};                  // persistent zero C for the second K-chain
    float hval[2][8];                      // tanh outputs (D slots), feed B + h_out
    float xv = xrow[0];                    // x pipeline, 1 step ahead

    for (int t = 0; t < T; ++t) {
        // ---- issue next x load + far prefetch first (off the critical path) ----
        const int tn = (t + 1 < T) ? (t + 1) : (T - 1);
        const float xv_next = xrow[tn];
        __builtin_prefetch(xrow + tn + 64, 0, 3);

        // ---- C = XP (mu-relabeled): one x scalar per lane per step (I == 1) ----
        v8f acc0, acc1;
#pragma unroll
        for (int r = 0; r < 8; ++r) {
            acc0[r] = fmaf(xv, wih[0][r], bia[0][r]);
            acc1[r] = fmaf(xv, wih[1][r], bia[1][r]);
        }
        v8f acc0b = zeroC, acc1b = zeroC;

        // ---- four independent 4-deep WMMA chains (2 M-tiles x 2 K-halves) ----
#pragma unroll
        for (int kk = 0; kk < 4; ++kk) {
            acc0 = __builtin_amdgcn_wmma_f32_16x16x4_f32(
                false, a_frag[0][kk], false, b_frag[kk], (short)0, acc0, false, false);
            acc1 = __builtin_amdgcn_wmma_f32_16x16x4_f32(
                false, a_frag[1][kk], false, b_frag[kk], (short)0, acc1, false, false);
            acc0b = __builtin_amdgcn_wmma_f32_16x16x4_f32(
                false, a_frag[0][kk + 4], false, b_frag[kk + 4], (short)0, acc0b, false, false);
            acc1b = __builtin_amdgcn_wmma_f32_16x16x4_f32(
                false, a_frag[1][kk + 4], false, b_frag[kk + 4], (short)0, acc1b, false, false);
        }
        acc0 = acc0 + acc0b;
        acc1 = acc1 + acc1b;

        // ---- tanh + fused O=1 output projection ----
        float p = 0.0f;
#pragma unroll
        for (int r = 0; r < 8; ++r) {
            hval[0][r] = fast_tanh(acc0[r]);
            hval[1][r] = fast_tanh(acc1[r]);
            p = fmaf(wfc[0][r], hval[0][r], p);
            p = fmaf(wfc[1][r], hval[1][r], p);
        }

        // ---- next-step B fragments: pure lane-local register rename ----
        // B slot q=4kk+2*half+j comes from D slot mu(q): mt=kk>>2,
        // r = ((kk&1)<<2) | (((kk>>1)&1)<<1) | j  -- consecutive pairs.
#pragma unroll
        for (int mt = 0; mt < 2; ++mt) {
            v2f f;
            f.x = hval[mt][0]; f.y = hval[mt][1]; b_frag[4 * mt + 0] = f;
            f.x = hval[mt][4]; f.y = hval[mt][5]; b_frag[4 * mt + 1] = f;
            f.x = hval[mt][2]; f.y = hval[mt][3]; b_frag[4 * mt + 2] = f;
            f.x = hval[mt][6]; f.y = hval[mt][7]; b_frag[4 * mt + 3] = f;
        }

        // both wave halves hold partials for the same batch column: fold halves
        p += __shfl_xor(p, 16, 32);
        if (lane < 16)
            orow[t] = p + bfc;                         // out[(b0+l15)*T + t]

        xv = xv_next;
    }

    // ---- final hidden state straight from registers (undo mu on the index) ----
#pragma unroll
    for (int mt = 0; mt < 2; ++mt)
#pragma unroll
        for (int r = 0; r < 8; ++r)
            h_out[(b0 + l15) * H + swap13(16 * mt + r + 8 * half)] = hval[mt][r];
}

extern "C" void kernel_launch(void* const* d_in, const int* in_sizes, int n_in,
                              void* d_out, int out_size, void* d_ws, size_t ws_size,
                              hipStream_t stream) {
    const float* x      = (const float*)d_in[0];   // [B, T, 1]
    const float* hidden = (const float*)d_in[1];   // [1, B, H]
    const float* W_ih   = (const float*)d_in[2];   // [H, 1]
    const float* W_hh   = (const float*)d_in[3];   // [H, H]
    const float* b_ih   = (const float*)d_in[4];   // [H]
    const float* b_hh   = (const float*)d_in[5];   // [H]
    const float* W_fc   = (const float*)d_in[6];   // [1, H]
    const float* b_fc   = (const float*)d_in[7];   // [1]

    const int H = 32;
    const int B = in_sizes[1] / H;                 // hidden flat = B*H
    const int T = in_sizes[0] / B;                 // x flat = B*T

    float* out   = (float*)d_out;                  // [B*T] then [B*H]
    float* h_out = out + (size_t)B * T;

    dim3 grid(B / 16), block(32);
    hipLaunchKernelGGL(rnn_scan_wmma_f32, grid, block, 0, stream,
                       x, hidden, W_ih, W_hh, b_ih, b_hh, W_fc, b_fc,
                       out, h_out, B, T);
}